// HeteroGCN_32014686224885
// MI455X (gfx1250) — compile-verified
//
#include <hip/hip_runtime.h>

// CDNA5 / gfx1250: wave32, WMMA 16x16x4 f32.
typedef __attribute__((ext_vector_type(2))) float v2f;
typedef __attribute__((ext_vector_type(8))) float v8f;

#define HDIM 64

// ---------------------------------------------------------------------------
// Scatter-sum pass: one (edge, 16B-chunk) per thread. Node tables + accum
// arrays are L2-resident (<< 192MB), so the f32 atomics resolve in L2.
// ---------------------------------------------------------------------------
__global__ __launch_bounds__(256)
void hg_scatter(const float* __restrict__ xsrc, const int* __restrict__ src,
                const int* __restrict__ dst, float* __restrict__ acc,
                float* __restrict__ cnt, int E)
{
  int idx = blockIdx.x * blockDim.x + threadIdx.x;
  int e = idx >> 4;
  if (e >= E) return;
  int c = (idx & 15) << 2;                 // float offset 0..60, step 4
  int s = src[e];
  int d = dst[e];
  const float4 v = *(const float4*)(xsrc + (size_t)s * HDIM + c);
  float* p = acc + (size_t)d * HDIM + c;
  atomicAdd(p + 0, v.x);
  atomicAdd(p + 1, v.y);
  atomicAdd(p + 2, v.z);
  atomicAdd(p + 3, v.w);
  if (c == 0) atomicAdd(cnt + d, 1.0f);    // edge count for the mean
}

// ---------------------------------------------------------------------------
// One fp32 GEMM product into four 16x16 accumulators:
//   C[16,64] += (inv * A[16,64]) @ Wsh^T   (Wsh is [64 out][64 in] in LDS)
// A frag per ISA 16x4 layout: lanes 0-15 -> K={k0,k0+1}, lanes 16-31 -> {k0+2,k0+3}.
// B frag B[k][n] = Wsh[n][k]: same lane-half striping over K.
// ---------------------------------------------------------------------------
__device__ __forceinline__ void mma_product(v8f cacc[4],
                                            const float* __restrict__ Abase,
                                            float inv,
                                            const float* __restrict__ Wsh,
                                            int m, int half)
{
#pragma unroll
  for (int k0 = 0; k0 < HDIM; k0 += 4) {
    const int kk = k0 + 2 * half;
    v2f a;
    a.x = Abase[kk] * inv;
    a.y = Abase[kk + 1] * inv;
#pragma unroll
    for (int nt = 0; nt < 4; ++nt) {
      v2f b = *(const v2f*)(Wsh + (size_t)(nt * 16 + m) * HDIM + kk);
      cacc[nt] = __builtin_amdgcn_wmma_f32_16x16x4_f32(
          false, a, false, b, (short)0, cacc[nt], false, false);
    }
  }
}

// ---------------------------------------------------------------------------
// Fused SAGE update for one dst node type (1 or 2 contributing edge types):
//   out = [relu]( meanA@WlA^T + [meanB@WlB^T] + x_dst@(WrA[+WrB])^T + bias )
// 8 waves/block, one 16-row tile per wave; wave-uniform exit keeps EXEC all-1s
// as required for WMMA.
// ---------------------------------------------------------------------------
__global__ __launch_bounds__(256)
void hg_sage(const float* __restrict__ accA, const float* __restrict__ cntA,
             const float* __restrict__ WlA, const float* __restrict__ blA,
             const float* __restrict__ WrA,
             const float* __restrict__ accB, const float* __restrict__ cntB,
             const float* __restrict__ WlB, const float* __restrict__ blB,
             const float* __restrict__ WrB,
             const float* __restrict__ xdst, float* __restrict__ out,
             int n_dst, int do_relu)
{
  __shared__ float sWlA[HDIM * HDIM];
  __shared__ float sWlB[HDIM * HDIM];
  __shared__ float sWr [HDIM * HDIM];
  __shared__ float sbias[HDIM];

  const int tid = threadIdx.x;
  const bool hasB = (accB != nullptr);

  // Stage weights to LDS; root weights of both edge types are pre-summed
  // (x @ Wr0^T + x @ Wr3^T == x @ (Wr0+Wr3)^T), saving a whole product.
  for (int i = tid; i < HDIM * HDIM; i += 256) {
    sWlA[i] = WlA[i];
    sWr[i]  = hasB ? (WrA[i] + WrB[i]) : WrA[i];
    sWlB[i] = hasB ? WlB[i] : 0.0f;
  }
  if (tid < HDIM) sbias[tid] = hasB ? (blA[tid] + blB[tid]) : blA[tid];
  __syncthreads();

  const int wave = tid >> 5;
  const int lane = tid & 31;
  const int tile = blockIdx.x * 8 + wave;
  if (tile * 16 >= n_dst) return;          // wave-uniform: EXEC stays full

  const int m = lane & 15;
  const int half = lane >> 4;
  const int row = tile * 16 + m;

  v8f cacc[4] = {};

  {
    const float invA = 1.0f / fmaxf(cntA[row], 1.0f);
    mma_product(cacc, accA + (size_t)row * HDIM, invA, sWlA, m, half);
  }
  if (hasB) {
    const float invB = 1.0f / fmaxf(cntB[row], 1.0f);
    mma_product(cacc, accB + (size_t)row * HDIM, invB, sWlB, m, half);
  }
  mma_product(cacc, xdst + (size_t)row * HDIM, 1.0f, sWr, m, half);

  // C/D layout: VGPR r holds row (r + 8*half), column = nt*16 + m.
#pragma unroll
  for (int nt = 0; nt < 4; ++nt) {
    const float bv = sbias[nt * 16 + m];
    float* obase = out + (size_t)(tile * 16) * HDIM + nt * 16 + m;
#pragma unroll
    for (int r = 0; r < 8; ++r) {
      float v = cacc[nt][r] + bv;
      if (do_relu) v = fmaxf(v, 0.0f);
      obase[(size_t)(r + 8 * half) * HDIM] = v;
    }
  }
}

// ---------------------------------------------------------------------------
extern "C" void kernel_launch(void* const* d_in, const int* in_sizes, int n_in,
                              void* d_out, int out_size, void* d_ws, size_t ws_size,
                              hipStream_t stream) {
  const int* src0 = (const int*)d_in[0]; const int* dst0 = (const int*)d_in[1];
  const int* src1 = (const int*)d_in[2]; const int* dst1 = (const int*)d_in[3];
  const int* src2 = (const int*)d_in[4]; const int* dst2 = (const int*)d_in[5];
  const int* src3 = (const int*)d_in[6]; const int* dst3 = (const int*)d_in[7];
  const float* ed  = (const float*)d_in[8];    // emb_drug    [Nd,64]
  const float* eg  = (const float*)d_in[9];    // emb_gene    [Ng,64]
  const float* es  = (const float*)d_in[10];   // emb_disease [Ns,64]
  const float* Wl0 = (const float*)d_in[11];
  const float* bl0 = (const float*)d_in[12];
  const float* Wr0 = (const float*)d_in[13];
  const float* Wl1 = (const float*)d_in[14];
  const float* bl1 = (const float*)d_in[15];
  const float* Wr1 = (const float*)d_in[16];

  const int E  = in_sizes[0];
  const int Nd = in_sizes[8]  / HDIM;
  const int Ng = in_sizes[9]  / HDIM;
  const int Ns = in_sizes[10] / HDIM;

  // Workspace carve-up (floats)
  float* ws = (float*)d_ws;
  size_t off = 0;
  float* acc0 = ws + off; off += (size_t)Ng * HDIM;   // drug->gene sums
  float* acc1 = ws + off; off += (size_t)Nd * HDIM;   // gene->drug sums
  float* acc2 = ws + off; off += (size_t)Ns * HDIM;   // gene->disease sums
  float* acc3 = ws + off; off += (size_t)Ng * HDIM;   // disease->gene sums
  float* cnt0 = ws + off; off += (size_t)Ng;
  float* cnt1 = ws + off; off += (size_t)Nd;
  float* cnt2 = ws + off; off += (size_t)Ns;
  float* cnt3 = ws + off; off += (size_t)Ng;
  const size_t accBytes = off * sizeof(float);
  float* xd1 = ws + off; off += (size_t)Nd * HDIM;    // layer-1 activations
  float* xg1 = ws + off; off += (size_t)Ng * HDIM;
  float* xs1 = ws + off; off += (size_t)Ns * HDIM;

  float* outd = (float*)d_out;
  float* outg = outd + (size_t)Nd * HDIM;
  float* outs = outg + (size_t)Ng * HDIM;

  const int sgrid = (E * 16 + 255) / 256;
  const dim3 blk(256);
  auto Wt = [](const float* base, int t) { return base + (size_t)t * HDIM * HDIM; };
  auto Bt = [](const float* base, int t) { return base + (size_t)t * HDIM; };
  const int gGene = (Ng / 16 + 7) / 8;
  const int gDrug = (Nd / 16 + 7) / 8;
  const int gDis  = (Ns / 16 + 7) / 8;

  // ---------------- Layer 1 ----------------
  hipMemsetAsync(ws, 0, accBytes, stream);
  hg_scatter<<<sgrid, blk, 0, stream>>>(ed, src0, dst0, acc0, cnt0, E);
  hg_scatter<<<sgrid, blk, 0, stream>>>(eg, src1, dst1, acc1, cnt1, E);
  hg_scatter<<<sgrid, blk, 0, stream>>>(eg, src2, dst2, acc2, cnt2, E);
  hg_scatter<<<sgrid, blk, 0, stream>>>(es, src3, dst3, acc3, cnt3, E);

  hg_sage<<<gGene, blk, 0, stream>>>(acc0, cnt0, Wt(Wl0,0), Bt(bl0,0), Wt(Wr0,0),
                                     acc3, cnt3, Wt(Wl0,3), Bt(bl0,3), Wt(Wr0,3),
                                     eg, xg1, Ng, 1);
  hg_sage<<<gDrug, blk, 0, stream>>>(acc1, cnt1, Wt(Wl0,1), Bt(bl0,1), Wt(Wr0,1),
                                     nullptr, nullptr, nullptr, nullptr, nullptr,
                                     ed, xd1, Nd, 1);
  hg_sage<<<gDis,  blk, 0, stream>>>(acc2, cnt2, Wt(Wl0,2), Bt(bl0,2), Wt(Wr0,2),
                                     nullptr, nullptr, nullptr, nullptr, nullptr,
                                     es, xs1, Ns, 1);

  // ---------------- Layer 2 ----------------
  hipMemsetAsync(ws, 0, accBytes, stream);
  hg_scatter<<<sgrid, blk, 0, stream>>>(xd1, src0, dst0, acc0, cnt0, E);
  hg_scatter<<<sgrid, blk, 0, stream>>>(xg1, src1, dst1, acc1, cnt1, E);
  hg_scatter<<<sgrid, blk, 0, stream>>>(xg1, src2, dst2, acc2, cnt2, E);
  hg_scatter<<<sgrid, blk, 0, stream>>>(xs1, src3, dst3, acc3, cnt3, E);

  hg_sage<<<gGene, blk, 0, stream>>>(acc0, cnt0, Wt(Wl1,0), Bt(bl1,0), Wt(Wr1,0),
                                     acc3, cnt3, Wt(Wl1,3), Bt(bl1,3), Wt(Wr1,3),
                                     xg1, outg, Ng, 0);
  hg_sage<<<gDrug, blk, 0, stream>>>(acc1, cnt1, Wt(Wl1,1), Bt(bl1,1), Wt(Wr1,1),
                                     nullptr, nullptr, nullptr, nullptr, nullptr,
                                     xd1, outd, Nd, 0);
  hg_sage<<<gDis,  blk, 0, stream>>>(acc2, cnt2, Wt(Wl1,2), Bt(bl1,2), Wt(Wr1,2),
                                     nullptr, nullptr, nullptr, nullptr, nullptr,
                                     xs1, outs, Ns, 0);
}